// Conv1DMod_73856257622622
// MI455X (gfx1250) — compile-verified
//
#include <hip/hip_runtime.h>
#include <hip/hip_bf16.h>
#include <math.h>
#include <stdint.h>

// ---------------------------------------------------------------------------
// Modulated conv1d (StyleGAN) on MI455X / gfx1250, wave32.
//   B=8, C=256, W=8192, F=256, K=3. out[b,f,x] = sum_{k,c} w[b,k,c,f]*feat[b,c,x+k-1]
// Roofline: 25.8 GFLOP vs 141 MB HBM (~6us floor). f32 WMMA (16x16x4) keeps
// reference fp32 precision on the matrix pipe. Feature slabs stream into LDS
// via gfx1250 async DMA (GLOBAL_LOAD_ASYNC_TO_LDS_B32 + s_wait_asynccnt),
// double-buffered to overlap the WMMA loop. Each A fragment feeds 8 WMMAs
// (8 position-subtiles per wave) to amortize weight reads from L2.
// ---------------------------------------------------------------------------

typedef __attribute__((ext_vector_type(2))) float v2f;
typedef __attribute__((ext_vector_type(8))) float v8f;

#define NB 8
#define NC 256
#define NW 8192
#define NF 256
#define NK 3
#define COEF 0.03608439182435161f /* 1/sqrt(3*256) */

// LDS feature slab: 64 channel rows x 132 cols (130 used: x0-1 .. x0+128),
// K-pair interleaved: element (r, col) -> ((r>>1)*COLSP + col)*2 + (r&1).
// COLSP=144 => row-pair stride 288 dwords == 32 mod 64 banks: the lanes16-31
// half of a b64 fragment read (rows +2) hits a disjoint 32-bank window.
#define COLSP 144
#define SLAB_DWORDS (32 * COLSP * 2) /* 9216 dwords = 36864 B per buffer */
#define FCOLS 132                    /* 64*132 = 33*256: uniform async count */
#define NFILL 33

__device__ __forceinline__ unsigned sidx(int r, int col) {
  return (((unsigned)(r >> 1) * COLSP + col) * 2) + (r & 1);
}

#define S_WAIT_ASYNCCNT(N) asm volatile("s_wait_asynccnt " N ::: "memory")

// Packed weight layout (floats): [B][K][F/16][C/4][lane=32][2]
//   ISA 16x4 f32 A-fragment: lanes 0-15 -> M=f%16, (VGPR0,VGPR1)=(K=c0,c0+1)
//                            lanes16-31 -> M=f%16, (VGPR0,VGPR1)=(K=c0+2,c0+3)
__device__ __forceinline__ size_t wpack_off(int b, int k, int fTile, int cChunk,
                                            int lane, int j) {
  return ((((size_t)b * NK + k) * (NF / 16) + fTile) * (NC / 4) + cChunk) * 64 +
         (size_t)lane * 2 + j;
}

// ---------------- Phase 1: modulate + demodulate + swizzle -----------------
__global__ __launch_bounds__(256) void modweights_kernel(
    const float* __restrict__ kern,   // [K, C, F]
    const float* __restrict__ style,  // [B, C]
    float* __restrict__ wPack) {
  __shared__ float red[256];
  const int b = blockIdx.x >> 8;  // F == 256
  const int f = blockIdx.x & 255;
  const int t = threadIdx.x;      // == c (C == 256); k handled by j-loop

  float wv[NK];
  float partial = 0.f;
  const float sm = style[b * NC + t] + 1.0f;
#pragma unroll
  for (int j = 0; j < NK; ++j) {
    float wm = kern[((size_t)j * NC + t) * NF + f] * COEF * sm;
    wv[j] = wm;
    partial += wm * wm;
  }
  red[t] = partial;
  __syncthreads();
#pragma unroll
  for (int s = 128; s > 0; s >>= 1) {
    if (t < s) red[t] += red[t + s];
    __syncthreads();
  }
  const float inv = rsqrtf(red[0]);

  const int m = f & 15, fTile = f >> 4;
  const int c = t;
  const int laneT = m + (((c & 3) >> 1) << 4);  // +16 when c%4 in {2,3}
  const int jj = c & 1;
#pragma unroll
  for (int j = 0; j < NK; ++j)
    wPack[wpack_off(b, j, fTile, c >> 2, laneT, jj)] = wv[j] * inv;
}

// Issue one chunk's DMA fill: 33 GLOBAL_LOAD_ASYNC_TO_LDS_B32 per wave, all
// lanes active. Per-thread LDS/global offsets are precomputed (chunk-invariant).
__device__ __forceinline__ void issue_fill(const char* gbase, unsigned ldsByteBase,
                                           const unsigned* loffA,
                                           const unsigned* goffA) {
#pragma unroll
  for (int i = 0; i < NFILL; ++i) {
    asm volatile("global_load_async_to_lds_b32 %0, %1, off"
                 :: "v"(ldsByteBase + loffA[i]),
                    "v"((unsigned long long)(uintptr_t)(gbase + goffA[i]))
                 : "memory");
  }
}

// ---------------- Phase 2: WMMA grouped conv -------------------------------
// Block: 256 thr (8 waves). Tile: 128 filters x 128 positions, one batch.
// Wave w -> filter tile w; 8 position subtiles (8 v8f accumulators), so each
// A fragment (1 global_load_b64) feeds 8 v_wmma_f32_16x16x4_f32.
__global__ __launch_bounds__(256) void conv_wmma_kernel(
    const float* __restrict__ feature,  // [B, C, W]
    const float* __restrict__ wPack,
    float* __restrict__ out) {          // [B, F, W]
  extern __shared__ float smem[];  // 2 * 36864 B = 73.7 KB of 320 KB WGP LDS

  const int b = blockIdx.z;
  const int fBlk = blockIdx.y;      // 0..1  (128 filters each)
  const int x0 = blockIdx.x * 128;  // 0..63 blocks
  const int tid = threadIdx.x;
  const int lane = tid & 31;
  const int wave = tid >> 5;

  const int fTile = fBlk * 8 + wave;  // 16-filter tile id (0..15)
  const int n = lane & 15;            // N (position) within WMMA tile
  const int ch = (lane >> 4) << 1;    // K half: 0 or 2

  v8f acc[8];
#pragma unroll
  for (int t = 0; t < 8; ++t) acc[t] = (v8f){};

  const float* featB = feature + (size_t)b * NC * NW;
  const unsigned ldsBase = (unsigned)(uintptr_t)(void*)smem;
  const bool leftEdge = (x0 == 0);
  const bool rightEdge = (x0 == NW - 128);

  // Chunk-invariant per-thread fill offsets (kept in VGPR arrays).
  unsigned loffA[NFILL], goffA[NFILL];
#pragma unroll
  for (int i = 0; i < NFILL; ++i) {
    const int idx = tid + i * 256;
    const int r = idx / FCOLS;
    const int col = idx - r * FCOLS;
    const int gx = min(max(x0 + col - 1, 0), NW - 1);  // clamp; zero-fix later
    goffA[i] = ((unsigned)r * NW + (unsigned)gx) * 4u;
    loffA[i] = sidx(r, col) * 4u;
  }

  // Prologue: start DMA of chunk 0 into buffer 0.
  issue_fill((const char*)featB, ldsBase, loffA, goffA);

  int buf = 0;
  for (int chunk = 0; chunk < 4; ++chunk) {
    const int c0 = chunk * 64;
    if (chunk < 3) {
      // Overlap: start next chunk's DMA into the other buffer, then wait only
      // for the current buffer's 33 ops (the 33 just issued stay in flight).
      issue_fill((const char*)featB + (size_t)(c0 + 64) * NW * 4,
                 ldsBase + (buf ^ 1) * (SLAB_DWORDS * 4), loffA, goffA);
      S_WAIT_ASYNCCNT("0x21");
    } else {
      S_WAIT_ASYNCCNT("0x0");
    }
    __syncthreads();  // all waves' DMA for `buf` complete & visible

    float* sbuf = smem + buf * SLAB_DWORDS;
    // SAME-padding fix-up (uniform branches; only the two edge blocks pay).
    if (leftEdge) {
      if (tid < 64) sbuf[sidx(tid, 0)] = 0.f;
      __syncthreads();
    }
    if (rightEdge) {
      if (tid < 64) sbuf[sidx(tid, 129)] = 0.f;
      __syncthreads();
    }

#pragma unroll
    for (int k = 0; k < NK; ++k) {
      const float* wBase =
          wPack + ((((size_t)b * NK + k) * (NF / 16) + fTile) * (NC / 4) +
                   (c0 >> 2)) * 64;
#pragma unroll 2
      for (int cc = 0; cc < 64; cc += 4) {
        // A fragment: coalesced 8B/lane from pre-swizzled weights (L2-resident)
        const v2f a = *(const v2f*)(wBase + (size_t)(cc >> 2) * 64 + lane * 2);
        const int rp = (cc + ch) >> 1;
        const float* sp = sbuf + ((unsigned)rp * COLSP + (n + k)) * 2;
#pragma unroll
        for (int t = 0; t < 8; ++t) {
          // B fragment: one conflict-free ds_load_b64 (K-pair interleaved slab)
          const v2f bf = *(const v2f*)(sp + t * 32);
          acc[t] = __builtin_amdgcn_wmma_f32_16x16x4_f32(
              false, a, false, bf, (short)0, acc[t], false, false);
        }
      }
    }
    __syncthreads();  // all reads of `buf` done before it is re-filled
    buf ^= 1;
  }

  // D layout: VGPR r -> M = r (+8 for lanes 16-31), N = lane&15
  const int fBase = fTile * 16 + ((lane >> 4) << 3);
  float* outB = out + ((size_t)b * NF + fBase) * NW + x0 + n;
#pragma unroll
  for (int t = 0; t < 8; ++t)
#pragma unroll
    for (int r = 0; r < 8; ++r)
      outB[(size_t)r * NW + t * 16] = acc[t][r];
}

// ---------------------------------------------------------------------------
extern "C" void kernel_launch(void* const* d_in, const int* in_sizes, int n_in,
                              void* d_out, int out_size, void* d_ws,
                              size_t ws_size, hipStream_t stream) {
  const float* feature = (const float*)d_in[0];  // [8,256,8192]
  const float* style = (const float*)d_in[1];    // [8,256]
  const float* kern = (const float*)d_in[2];     // [3,256,256]
  float* out = (float*)d_out;                    // [8,256,8192]
  float* wPack = (float*)d_ws;                   // 8*3*256*256*4B = 6.29 MB

  modweights_kernel<<<dim3(NB * NF), 256, 0, stream>>>(kern, style, wPack);

  dim3 grid(NW / 128, NF / 128, NB);  // 64 x 2 x 8
  conv_wmma_kernel<<<grid, 256, 2 * SLAB_DWORDS * 4, stream>>>(feature, wPack,
                                                               out);
}